// GeoLSTLayer_38689065402612
// MI455X (gfx1250) — compile-verified
//
#include <hip/hip_runtime.h>
#include <math.h>

// ---------------------------------------------------------------------------
// GVP-GNN attention layer for MI455X (gfx1250, wave32).
// Heavy linear layers run on V_WMMA_F32_16X16X4_F32 (exact fp32 semantics).
// Structure:
//   K1 node_gvp_kernel      : q,k,v GVPs for all nodes (WMMA, K=144)
//   K2 edge_energy_kernel   : ek GVP (WMMA, K=48) fused with attention energy
//                             + scatter-max (ordered-uint atomicMax)
//   K3 edge_scatter_kernel  : ev GVP (WMMA, K=48) fused with exp/denom and
//                             atomic scatter-add of messages
//   K4 node_gvp_kernel      : final GVP on (accum/denom) -> d_out
// ---------------------------------------------------------------------------

constexpr int kN  = 20000;          // nodes
constexpr int kE  = 400000;         // edges
constexpr int kSI = 128;            // node scalar dim
constexpr int kVI = 16;             // node vector dim
constexpr int kSO = 128;            // out scalar dim
constexpr int kVO = 16;             // out vector dim
constexpr int kES = 32;             // edge scalar dim
constexpr int kH  = 4;              // heads
constexpr int kHS = 32;             // scalar per head
constexpr int kHV = 4;              // vector per head
constexpr int kKN = kSI + kVI;      // 144 : node GVP concat dim
constexpr int kKE = kES + kVO;      // 48  : edge GVP concat dim (h=max(1,16)=16)
constexpr float kInvScale = 1.0f / 6.0f;  // 1/sqrt(HS+HV)

static_assert(kN % 16 == 0 && kE % 16 == 0, "tile divisibility");

typedef __attribute__((ext_vector_type(2))) float v2f;
typedef __attribute__((ext_vector_type(8))) float v8f;

struct Gvp {
  const float* wh; const float* ws_w; const float* ws_b;
  const float* wv; const float* wsv_w; const float* wsv_b;
};
struct NodeLaunch {
  Gvp    g[3];
  float* out_s[3];
  float* out_v[3];
};

__device__ __forceinline__ float sigm(float x) { return 1.0f / (1.0f + expf(-x)); }

// order-preserving float<->uint key for atomicMax-based segment max
__device__ __forceinline__ unsigned fkey(float f) {
  unsigned u = __float_as_uint(f);
  return (u & 0x80000000u) ? ~u : (u | 0x80000000u);
}
__device__ __forceinline__ float unfkey(unsigned k) {
  unsigned u = (k & 0x80000000u) ? (k ^ 0x80000000u) : ~k;
  return __uint_as_float(u);
}

// One 16x16 output tile of  D = X[16xK] * ws_w^T[Kx16]  using fp32 WMMA.
// A fragment from LDS (row = lane&15, K split 0..1 / 2..3 across half-waves),
// B fragment straight from global ws_w (row o = ob+lane&15, element k+koff).
template <int K, int PX>
__device__ __forceinline__ v8f wmma_rowtile(float (*X)[PX],
                                            const float* __restrict__ wsw,
                                            int ob, int lane) {
  const int nn   = lane & 15;
  const int koff = (lane >> 4) << 1;
  const float* wrow = wsw + (size_t)(ob + nn) * K;
  v8f acc = {0.f, 0.f, 0.f, 0.f, 0.f, 0.f, 0.f, 0.f};
#pragma unroll
  for (int k = 0; k < K; k += 4) {
    v2f a, b;
    a.x = X[nn][k + koff];
    a.y = X[nn][k + koff + 1];
    b.x = wrow[k + koff];
    b.y = wrow[k + koff + 1];
    acc = __builtin_amdgcn_wmma_f32_16x16x4_f32(false, a, false, b, (short)0,
                                                acc, false, false);
  }
  return acc;
}

// ------------------------- node GVP (also final GVP) -----------------------
// block: 256 thr (8 waves) = 16 nodes; wave w owns output cols [16w,16w+16).
__global__ __launch_bounds__(256)
void node_gvp_kernel(const float* __restrict__ in_s,
                     const float* __restrict__ in_v,
                     const float* __restrict__ denom,   // nullable
                     NodeLaunch P) {
  __shared__ float X[16][145];          // [s(128) | vn(16)], odd pitch
  __shared__ float xvb[16][kVI][3];
  __shared__ float vhb[16][16][3];
  __shared__ float sgb[16][kSO];        // sigmoid(s_lin)

  const int tid = threadIdx.x;
  const int lane = tid & 31, wave = tid >> 5;
  const int n0 = blockIdx.x * 16;
  const Gvp g = P.g[blockIdx.y];

  for (int i = tid; i < 16 * kSI; i += 256) {
    int r = i >> 7, c = i & 127;
    float v = in_s[(size_t)(n0 + r) * kSI + c];
    if (denom) v *= 1.0f / fmaxf(denom[(size_t)(n0 + r) * kH + (c >> 5)], 1e-16f);
    X[r][c] = v;
  }
  for (int i = tid; i < 16 * kVI * 3; i += 256) {
    int r = i / (kVI * 3), rem = i % (kVI * 3);
    float v = in_v[(size_t)n0 * kVI * 3 + i];
    if (denom) v *= 1.0f / fmaxf(denom[(size_t)(n0 + r) * kH + ((rem / 3) >> 2)], 1e-16f);
    xvb[r][rem / 3][rem % 3] = v;
  }
  __syncthreads();

  {  // vh = wh @ v ; vn = ||vh||
    int r = tid >> 4, h = tid & 15;
    const float* whr = g.wh + h * kVI;
    float a0 = 0.f, a1 = 0.f, a2 = 0.f;
#pragma unroll
    for (int v = 0; v < kVI; ++v) {
      float w = whr[v];
      a0 += w * xvb[r][v][0]; a1 += w * xvb[r][v][1]; a2 += w * xvb[r][v][2];
    }
    vhb[r][h][0] = a0; vhb[r][h][1] = a1; vhb[r][h][2] = a2;
    X[r][kSI + h] = sqrtf(fmaxf(a0 * a0 + a1 * a1 + a2 * a2, 1e-8f));
  }
  __syncthreads();

  const int ob = wave * 16;
  v8f acc = wmma_rowtile<kKN, 145>(X, g.ws_w, ob, lane);

  float* outS = P.out_s[blockIdx.y];
  const int nn = lane & 15;
  const int rb = (lane >> 4) ? 8 : 0;
#pragma unroll
  for (int i = 0; i < 8; ++i) {
    int r = i + rb, o = ob + nn;
    float s  = acc[i] + g.ws_b[o];
    float sg = sigm(s);
    sgb[r][o] = sg;
    outS[(size_t)(n0 + r) * kSO + o] = s * sg;   // silu
  }
  __syncthreads();

  {  // gate = wsv(sigmoid(s)) ; v_out = sigmoid(gate) * (wv @ vh)
    int r = tid >> 4, o2 = tid & 15;
    const float* wr = g.wsv_w + o2 * kSO;
    float ga = g.wsv_b[o2];
    for (int j = 0; j < kSO; ++j) ga += sgb[r][j] * wr[j];
    float gs = sigm(ga);
    const float* wvr = g.wv + o2 * 16;
    float c0 = 0.f, c1 = 0.f, c2 = 0.f;
#pragma unroll
    for (int h = 0; h < 16; ++h) {
      float w = wvr[h];
      c0 += w * vhb[r][h][0]; c1 += w * vhb[r][h][1]; c2 += w * vhb[r][h][2];
    }
    float* outV = P.out_v[blockIdx.y];
    size_t b = ((size_t)(n0 + r) * kVO + o2) * 3;
    outV[b + 0] = c0 * gs; outV[b + 1] = c1 * gs; outV[b + 2] = c2 * gs;
  }
}

// ------------------------- shared edge-GVP body ----------------------------
__device__ __forceinline__ void edge_gvp(int e0, int tid, int lane, int wave,
    const float* __restrict__ edge_s, const float* __restrict__ edge_v,
    const Gvp& g, float (*Xe)[49], float (*evh)[16][3],
    float (*esl)[kSO], float (*esg)[kSO], float (*evv)[16][3]) {
  for (int i = tid; i < 16 * kES; i += 256) {
    int r = i >> 5, c = i & 31;
    Xe[r][c] = edge_s[(size_t)(e0 + r) * kES + c];
  }
  {  // vh[h] = wh[h] * edge_v (vi = 1)
    int r = tid >> 4, h = tid & 15;
    float w = g.wh[h];
    size_t b = (size_t)(e0 + r) * 3;
    float a0 = w * edge_v[b], a1 = w * edge_v[b + 1], a2 = w * edge_v[b + 2];
    evh[r][h][0] = a0; evh[r][h][1] = a1; evh[r][h][2] = a2;
    Xe[r][kES + h] = sqrtf(fmaxf(a0 * a0 + a1 * a1 + a2 * a2, 1e-8f));
  }
  __syncthreads();

  const int ob = wave * 16;
  v8f acc = wmma_rowtile<kKE, 49>(Xe, g.ws_w, ob, lane);

  const int nn = lane & 15;
  const int rb = (lane >> 4) ? 8 : 0;
#pragma unroll
  for (int i = 0; i < 8; ++i) {
    int r = i + rb, o = ob + nn;
    float s = acc[i] + g.ws_b[o];
    esl[r][o] = s;
    esg[r][o] = sigm(s);
  }
  __syncthreads();

  {
    int r = tid >> 4, o2 = tid & 15;
    const float* wr = g.wsv_w + o2 * kSO;
    float ga = g.wsv_b[o2];
    for (int j = 0; j < kSO; ++j) ga += esg[r][j] * wr[j];
    float gs = sigm(ga);
    const float* wvr = g.wv + o2 * 16;
    float c0 = 0.f, c1 = 0.f, c2 = 0.f;
#pragma unroll
    for (int h = 0; h < 16; ++h) {
      float w = wvr[h];
      c0 += w * evh[r][h][0]; c1 += w * evh[r][h][1]; c2 += w * evh[r][h][2];
    }
    evv[r][o2][0] = c0 * gs; evv[r][o2][1] = c1 * gs; evv[r][o2][2] = c2 * gs;
  }
  __syncthreads();
}

// ------------------------- pass A: energies + segment max ------------------
__global__ __launch_bounds__(256)
void edge_energy_kernel(const float* __restrict__ edge_s,
                        const float* __restrict__ edge_v, Gvp g,
                        const float* __restrict__ q_s, const float* __restrict__ q_v,
                        const float* __restrict__ k_s, const float* __restrict__ k_v,
                        const int* __restrict__ ei,
                        float* __restrict__ energy, unsigned* __restrict__ emaxk) {
  __shared__ float Xe[16][49];
  __shared__ float evh[16][16][3];
  __shared__ float esl[16][kSO];
  __shared__ float esg[16][kSO];
  __shared__ float evv[16][16][3];
  const int tid = threadIdx.x, lane = tid & 31, wave = tid >> 5;
  const int e0 = blockIdx.x * 16;
  edge_gvp(e0, tid, lane, wave, edge_s, edge_v, g, Xe, evh, esl, esg, evv);

  if (tid < 64) {
    int el = tid >> 2, h = tid & 3;
    int e = e0 + el;
    int s = ei[e], d = ei[kE + e];
    float en = 0.f;
    const float* qs = q_s + (size_t)d * kSO + h * kHS;
    const float* ks = k_s + (size_t)s * kSO + h * kHS;
#pragma unroll 4
    for (int j = 0; j < kHS; ++j) {
      int o = h * kHS + j;
      en += qs[j] * (ks[j] + esl[el][o] * esg[el][o]);  // silu(ek_s)
    }
    const float* qv = q_v + ((size_t)d * kVO + h * kHV) * 3;
    const float* kv = k_v + ((size_t)s * kVO + h * kHV) * 3;
#pragma unroll
    for (int t = 0; t < kHV * 3; ++t) {
      en += qv[t] * (kv[t] + evv[el][h * kHV + t / 3][t % 3]);
    }
    en *= kInvScale;
    energy[(size_t)e * kH + h] = en;
    atomicMax(&emaxk[(size_t)d * kH + h], fkey(en));
  }
}

// ------------------------- pass B: softmax + scatter -----------------------
__global__ __launch_bounds__(256)
void edge_scatter_kernel(const float* __restrict__ edge_s,
                         const float* __restrict__ edge_v, Gvp g,
                         const float* __restrict__ v_s, const float* __restrict__ v_v,
                         const int* __restrict__ ei,
                         const float* __restrict__ energy,
                         const unsigned* __restrict__ emaxk,
                         float* __restrict__ denom,
                         float* __restrict__ acc_s, float* __restrict__ acc_v) {
  __shared__ float Xe[16][49];
  __shared__ float evh[16][16][3];
  __shared__ float esl[16][kSO];
  __shared__ float esg[16][kSO];
  __shared__ float evv[16][16][3];
  __shared__ float exb[16][kH];
  __shared__ int srcb[16], dstb[16];
  const int tid = threadIdx.x, lane = tid & 31, wave = tid >> 5;
  const int e0 = blockIdx.x * 16;
  edge_gvp(e0, tid, lane, wave, edge_s, edge_v, g, Xe, evh, esl, esg, evv);

  if (tid < 16) { srcb[tid] = ei[e0 + tid]; dstb[tid] = ei[kE + e0 + tid]; }
  __syncthreads();
  if (tid < 64) {
    int el = tid >> 2, h = tid & 3;
    int e = e0 + el, d = dstb[el];
    float ex = expf(energy[(size_t)e * kH + h] - unfkey(emaxk[(size_t)d * kH + h]));
    exb[el][h] = ex;
    atomicAdd(&denom[(size_t)d * kH + h], ex);
  }
  __syncthreads();
  for (int i = tid; i < 16 * kSO; i += 256) {   // scalar messages
    int el = i >> 7, o = i & 127, h = o >> 5;
    float m = exb[el][h] * (v_s[(size_t)srcb[el] * kSO + o] + esl[el][o] * esg[el][o]);
    atomicAdd(&acc_s[(size_t)dstb[el] * kSO + o], m);
  }
  for (int i = tid; i < 16 * kVO * 3; i += 256) {  // vector messages
    int el = i / 48, r = i % 48, o2 = r / 3, c = r % 3, h = o2 >> 2;
    float m = exb[el][h] * (v_v[((size_t)srcb[el] * kVO + o2) * 3 + c] + evv[el][o2][c]);
    atomicAdd(&acc_v[((size_t)dstb[el] * kVO + o2) * 3 + c], m);
  }
}

// ---------------------------------------------------------------------------
extern "C" void kernel_launch(void* const* d_in, const int* in_sizes, int n_in,
                              void* d_out, int out_size, void* d_ws, size_t ws_size,
                              hipStream_t stream) {
  const float *x_s = nullptr, *x_v = nullptr, *edge_s = nullptr, *edge_v = nullptr;
  const int* edge_index = nullptr;
  Gvp G[6];  // q, k, v, ek, ev, out

  if (n_in >= 41) {
    // Each params leaf is its own input. Detect top-level / group / leaf order
    // (insertion vs alphabetical pytree flattening) from in_sizes.
    int p0, ix_xs, ix_xv, ix_es, ix_ev, ix_ei;
    if (in_sizes[0] == kN * kSI) {            // insertion: x_s,x_v,e_s,e_v,params,e_idx
      ix_xs = 0; ix_xv = 1; ix_es = 2; ix_ev = 3; p0 = 4; ix_ei = n_in - 1;
    } else {                                  // alphabetical: e_idx,e_s,e_v,params,x_s,x_v
      ix_ei = 0; ix_es = 1; ix_ev = 2; p0 = 3; ix_xs = n_in - 2; ix_xv = n_in - 1;
    }
    const bool inner_alpha = (in_sizes[p0 + 1] == kSO);  // ws_b(128) second => alpha
    Gvp tmp[6]; bool node0 = false;
    for (int gi = 0; gi < 6; ++gi) {
      int b = p0 + gi * 6;
      const float* L[6];
      for (int j = 0; j < 6; ++j) L[j] = (const float*)d_in[b + j];
      Gvp q;
      if (inner_alpha) { q.wh = L[0]; q.ws_b = L[1]; q.ws_w = L[2];
                         q.wsv_b = L[3]; q.wsv_w = L[4]; q.wv = L[5]; }
      else             { q.wh = L[0]; q.ws_w = L[1]; q.ws_b = L[2];
                         q.wv = L[3]; q.wsv_w = L[4]; q.wsv_b = L[5]; }
      tmp[gi] = q;
      if (gi == 0) node0 = (in_sizes[b + (inner_alpha ? 2 : 1)] == kSO * kKN);
    }
    if (node0) {  // insertion: q,k,v,ek,ev,out
      G[0] = tmp[0]; G[1] = tmp[1]; G[2] = tmp[2];
      G[3] = tmp[3]; G[4] = tmp[4]; G[5] = tmp[5];
    } else {      // alphabetical: ek,ev,k,out,q,v
      G[3] = tmp[0]; G[4] = tmp[1]; G[1] = tmp[2];
      G[5] = tmp[3]; G[0] = tmp[4]; G[2] = tmp[5];
    }
    x_s = (const float*)d_in[ix_xs];  x_v = (const float*)d_in[ix_xv];
    edge_s = (const float*)d_in[ix_es]; edge_v = (const float*)d_in[ix_ev];
    edge_index = (const int*)d_in[ix_ei];
  } else {
    // params packed into a single buffer at d_in[4], insertion leaf order.
    x_s = (const float*)d_in[0]; x_v = (const float*)d_in[1];
    edge_s = (const float*)d_in[2]; edge_v = (const float*)d_in[3];
    edge_index = (const int*)d_in[5];
    const float* b = (const float*)d_in[4];
    auto mk = [&b](bool node) -> Gvp {
      Gvp q; int k = (node ? kSI : kES) + 16;
      q.wh = b;    b += node ? 16 * 16 : 16;
      q.ws_w = b;  b += kSO * k;
      q.ws_b = b;  b += kSO;
      q.wv = b;    b += kVO * 16;
      q.wsv_w = b; b += kVO * kSO;
      q.wsv_b = b; b += kVO;
      return q;
    };
    G[0] = mk(true); G[1] = mk(true); G[2] = mk(true);
    G[3] = mk(false); G[4] = mk(false); G[5] = mk(true);
  }

  // ---- workspace carve (~63.4 MB of floats) ----
  float* w = (float*)d_ws;
  auto take = [&w](size_t n) { float* p = w; w += n; return p; };
  float* q_s = take((size_t)kN * kSO);
  float* k_s = take((size_t)kN * kSO);
  float* v_s = take((size_t)kN * kSO);
  float* q_v = take((size_t)kN * kVO * 3);
  float* k_v = take((size_t)kN * kVO * 3);
  float* v_v = take((size_t)kN * kVO * 3);
  float* energy = take((size_t)kE * kH);
  unsigned* emaxk = (unsigned*)take((size_t)kN * kH);
  float* denom = take((size_t)kN * kH);
  float* acc_s = take((size_t)kN * kSO);
  float* acc_v = take((size_t)kN * kVO * 3);

  // zero emaxk (key 0 == -inf sentinel), denom, accumulators — every call
  size_t zero_floats = (size_t)kN * kH * 2 + (size_t)kN * kSO + (size_t)kN * kVO * 3;
  hipMemsetAsync((void*)emaxk, 0, zero_floats * sizeof(float), stream);

  NodeLaunch P1;
  P1.g[0] = G[0]; P1.g[1] = G[1]; P1.g[2] = G[2];
  P1.out_s[0] = q_s; P1.out_s[1] = k_s; P1.out_s[2] = v_s;
  P1.out_v[0] = q_v; P1.out_v[1] = k_v; P1.out_v[2] = v_v;
  node_gvp_kernel<<<dim3(kN / 16, 3), 256, 0, stream>>>(x_s, x_v, nullptr, P1);

  edge_energy_kernel<<<kE / 16, 256, 0, stream>>>(edge_s, edge_v, G[3],
      q_s, q_v, k_s, k_v, edge_index, energy, emaxk);

  edge_scatter_kernel<<<kE / 16, 256, 0, stream>>>(edge_s, edge_v, G[4],
      v_s, v_v, edge_index, energy, emaxk, denom, acc_s, acc_v);

  NodeLaunch P2;
  P2.g[0] = G[5]; P2.g[1] = G[5]; P2.g[2] = G[5];
  float* out = (float*)d_out;
  P2.out_s[0] = out;                      P2.out_v[0] = out + (size_t)kN * kSO;
  P2.out_s[1] = P2.out_s[0]; P2.out_s[2] = P2.out_s[0];
  P2.out_v[1] = P2.out_v[0]; P2.out_v[2] = P2.out_v[0];
  node_gvp_kernel<<<dim3(kN / 16, 1), 256, 0, stream>>>(acc_s, acc_v, denom, P2);
}